// AugmentPipe_8332236554712
// MI455X (gfx1250) — compile-verified
//
#include <hip/hip_runtime.h>

typedef __attribute__((ext_vector_type(2))) float v2f;
typedef __attribute__((ext_vector_type(8))) float v8f;

#define WIN  256
#define WPAD 766
#define WUP  1532
#define WO   524
#define PI_F 3.14159265358979323846f

// workspace layout (float offsets)
#define OFF_THETA 0                       // 8 * 6
#define OFF_WYU   64                      // 16x16  upsample vertical band (A), gain 4
#define OFF_WXU   320                     // 16x16  upsample horizontal band (B)
#define OFF_WDA   576                     // 16x48  downsample vertical band (A)
#define OFF_WDB   1344                    // 48x16  downsample horizontal band (B)
#define OFF_UP    4096                    // 3 * 1532 * 1532
#define OFF_GS    (4096 + 3 * WUP * WUP)  // 3 * 524 * 524

// ---------------- 3x3 matrix helpers (per-image affine composition) --------
__device__ inline void mm3(float* o, const float* a, const float* b) {
#pragma unroll
  for (int i = 0; i < 3; i++)
#pragma unroll
    for (int j = 0; j < 3; j++)
      o[i * 3 + j] = a[i * 3 + 0] * b[0 + j] + a[i * 3 + 1] * b[3 + j] + a[i * 3 + 2] * b[6 + j];
}
__device__ inline void rmul(float* g, const float* m) {  // g = g @ m
  float t[9]; mm3(t, g, m);
#pragma unroll
  for (int i = 0; i < 9; i++) g[i] = t[i];
}
__device__ inline void lmul(float* g, const float* m) {  // g = m @ g
  float t[9]; mm3(t, m, g);
#pragma unroll
  for (int i = 0; i < 9; i++) g[i] = t[i];
}
__device__ inline void setS(float* m, float sx, float sy) {
  m[0] = sx; m[1] = 0; m[2] = 0; m[3] = 0; m[4] = sy; m[5] = 0; m[6] = 0; m[7] = 0; m[8] = 1;
}
__device__ inline void setT(float* m, float tx, float ty) {
  m[0] = 1; m[1] = 0; m[2] = tx; m[3] = 0; m[4] = 1; m[5] = ty; m[6] = 0; m[7] = 0; m[8] = 1;
}
__device__ inline void setR(float* m, float th) {
  float c = __cosf(th), s = __sinf(th);
  m[0] = c; m[1] = -s; m[2] = 0; m[3] = s; m[4] = c; m[5] = 0; m[6] = 0; m[7] = 0; m[8] = 1;
}

// ---------------- init: per-image affine + filter band matrices ------------
__global__ void init_kernel(const float* __restrict__ u_flip, const float* __restrict__ u_rot90,
                            const float* __restrict__ u_int, const float* __restrict__ z_scale,
                            const float* __restrict__ u_rot1, const float* __restrict__ z_aniso,
                            const float* __restrict__ u_rot2, const float* __restrict__ z_frac,
                            const float* __restrict__ f, float* __restrict__ ws) {
  const int tid = threadIdx.x;

  if (tid < 8) {
    const int n = tid;
    float G[9], M[9];
    float i_f = floorf(u_flip[n] * 2.f);
    setS(G, 1.f / (1.f - 2.f * i_f), 1.f);
    float i_r = floorf(u_rot90[n] * 4.f);
    setR(M, 1.5707963267948966f * i_r); rmul(G, M);
    float t0 = (u_int[2 * n + 0] * 2.f - 1.f) * 0.125f;
    float t1 = (u_int[2 * n + 1] * 2.f - 1.f) * 0.125f;
    setT(M, -rintf(t0 * (float)WIN), -rintf(t1 * (float)WIN)); rmul(G, M);
    float s = exp2f(z_scale[n] * 0.2f);
    setS(M, 1.f / s, 1.f / s); rmul(G, M);
    setR(M, (u_rot1[n] * 2.f - 1.f) * PI_F); rmul(G, M);
    float sa = exp2f(z_aniso[n] * 0.2f);
    setS(M, 1.f / sa, sa); rmul(G, M);
    setR(M, (u_rot2[n] * 2.f - 1.f) * PI_F); rmul(G, M);
    setT(M, -z_frac[2 * n + 0] * 0.125f * (float)WIN,
            -z_frac[2 * n + 1] * 0.125f * (float)WIN); rmul(G, M);
    // G = CS(2,2) @ G @ CS(.5,.5); G = CT(-.5,-.5) @ G @ CT(.5,.5)
    setS(M, 0.5f, 0.5f); rmul(G, M); setS(M, 2.f, 2.f); lmul(G, M);
    setT(M, 0.5f, 0.5f); rmul(G, M); setT(M, -0.5f, -0.5f); lmul(G, M);
    // G = CS(2/Wup,2/Hup) @ G @ CS(Wo/2,Ho/2)
    setS(M, (float)WO * 0.5f, (float)WO * 0.5f); rmul(G, M);
    setS(M, 2.f / (float)WUP, 2.f / (float)WUP); lmul(G, M);

    const float t00 = G[0], t01 = G[1], t02 = G[2];
    const float t10 = G[3], t11 = G[4], t12 = G[5];
    // fold grid-sample normalization: ix = Ax*ox + Bx*oy + Cx (ups coords)
    const float Ax = t00 * (float)WUP / (float)WO;
    const float Bx = t01 * (float)WUP / (float)WO;
    const float Cx = (t00 * (1.f / (float)WO - 1.f) + t01 * (1.f / (float)WO - 1.f) + t02 + 1.f)
                     * ((float)WUP * 0.5f) - 0.5f;
    const float Ay = t10 * (float)WUP / (float)WO;
    const float By = t11 * (float)WUP / (float)WO;
    const float Cy = (t10 * (1.f / (float)WO - 1.f) + t11 * (1.f / (float)WO - 1.f) + t12 + 1.f)
                     * ((float)WUP * 0.5f) - 0.5f;
    float* th = ws + OFF_THETA + n * 6;
    th[0] = Ax; th[1] = Bx; th[2] = Cx; th[3] = Ay; th[4] = By; th[5] = Cy;
  }

  // Upsample bands: fz = reverse(f); vertical has gain 4.
  {
    int m = tid >> 4, k = tid & 15;          // WyU[m][k]
    int j = 2 * k - m;
    ws[OFF_WYU + tid] = (j >= 0 && j < 12) ? f[11 - j] * 4.f : 0.f;
    int kb = tid >> 4, nb = tid & 15;        // WxU[k][n]
    j = 2 * kb - nb;
    ws[OFF_WXU + tid] = (j >= 0 && j < 12) ? f[11 - j] : 0.f;
  }
  // Downsample bands: fz = f, gain 1.
  for (int t = tid; t < 16 * 48; t += 256) {
    int m = t / 48, r = t % 48;              // WdA[m][r]
    int j = r - 2 * m;
    ws[OFF_WDA + t] = (j >= 0 && j < 12) ? f[j] : 0.f;
  }
  for (int t = tid; t < 48 * 16; t += 256) {
    int k = t / 16, nb = t % 16;             // WdB[k][n]
    int j = k - 2 * nb;
    ws[OFF_WDB + t] = (j >= 0 && j < 12) ? f[j] : 0.f;
  }
}

// ---------------- branchless reflect-padded read of the 256^2 source -------
// padded index p in [0,766) maps to source via single reflection about the
// borders; out-of-range p (dilation zero-pad region) contributes 0.
// reflect(i) for i = p-255 in [-255,510] is min(|i|, 510-|i|), done on the
// clamped index so the gather address is always valid; validity folds into a
// select on the loaded value (no EXEC divergence around the load).
__device__ inline float load_pad(const float* __restrict__ s, int py, int px) {
  const bool ok = ((unsigned)py <= 765u) & ((unsigned)px <= 765u);
  int pyc = py < 0 ? 0 : (py > 765 ? 765 : py);
  int pxc = px < 0 ? 0 : (px > 765 ? 765 : px);
  int ay = pyc - 255; ay = ay < 0 ? -ay : ay;
  int ax = pxc - 255; ax = ax < 0 ? -ax : ax;
  const int iy = ay < 510 - ay ? ay : 510 - ay;
  const int ix = ax < 510 - ax ? ax : 510 - ax;
  const float v = s[iy * WIN + ix];
  return ok ? v : 0.f;
}

// ---------------- upsample x2 (reflect pad fused), 16x16 tile via WMMA -----
__global__ void __launch_bounds__(32)
upsample_kernel(const float* __restrict__ img, const float* __restrict__ ws,
                float* __restrict__ up) {
  const int lane = threadIdx.x, lo = lane & 15, hi = lane >> 4;
  const int c = blockIdx.z;
  const float* src = img + c * WIN * WIN;
  const float* WyU = ws + OFF_WYU;
  const float* WxU = ws + OFF_WXU;
  const int Y0 = blockIdx.y * 16, X0 = blockIdx.x * 16;
  const int tyb = (Y0 >> 1) - 3, txb = (X0 >> 1) - 3;

  __builtin_prefetch(src + ((tyb < 0 ? 0 : tyb) & 255) * WIN, 0, 0);

  __shared__ float V[16][17];
  v8f acc = {0.f, 0.f, 0.f, 0.f, 0.f, 0.f, 0.f, 0.f};
#pragma unroll
  for (int kk = 0; kk < 4; kk++) {
    const int k0 = 4 * kk + 2 * hi;
    v2f a, b;
    a.x = WyU[lo * 16 + k0];
    a.y = WyU[lo * 16 + k0 + 1];
    b.x = load_pad(src, tyb + k0, txb + lo);
    b.y = load_pad(src, tyb + k0 + 1, txb + lo);
    acc = __builtin_amdgcn_wmma_f32_16x16x4_f32(false, a, false, b, (short)0, acc, false, false);
  }
#pragma unroll
  for (int v = 0; v < 8; v++) V[v + 8 * hi][lo] = acc[v];
  __syncthreads();

  v8f o = {0.f, 0.f, 0.f, 0.f, 0.f, 0.f, 0.f, 0.f};
#pragma unroll
  for (int kk = 0; kk < 4; kk++) {
    const int k0 = 4 * kk + 2 * hi;
    v2f a, b;
    a.x = V[lo][k0];
    a.y = V[lo][k0 + 1];
    b.x = WxU[k0 * 16 + lo];
    b.y = WxU[(k0 + 1) * 16 + lo];
    o = __builtin_amdgcn_wmma_f32_16x16x4_f32(false, a, false, b, (short)0, o, false, false);
  }
#pragma unroll
  for (int v = 0; v < 8; v++) {
    const int Y = Y0 + v + 8 * hi, X = X0 + lo;
    if (Y < WUP && X < WUP) up[c * WUP * WUP + Y * WUP + X] = o[v];
  }
}

// ---------------- affine bilinear grid sample (branchless taps) ------------
__device__ inline float smp_up(const float* __restrict__ P, int yy, int xx) {
  const bool ok = ((unsigned)yy < (unsigned)WUP) & ((unsigned)xx < (unsigned)WUP);
  const int yc = yy < 0 ? 0 : (yy > WUP - 1 ? WUP - 1 : yy);
  const int xc = xx < 0 ? 0 : (xx > WUP - 1 ? WUP - 1 : xx);
  const float v = P[yc * WUP + xc];
  return ok ? v : 0.f;
}

__global__ void gridsample_kernel(const float* __restrict__ up, const float* __restrict__ th,
                                  float* __restrict__ gs) {
  const int idx = blockIdx.x * blockDim.x + threadIdx.x;
  if (idx >= 3 * WO * WO) return;
  const int c = idx / (WO * WO);
  const int r = idx - c * WO * WO;
  const int oy = r / WO, ox = r - oy * WO;
  const float Ax = th[0], Bx = th[1], Cx = th[2], Ay = th[3], By = th[4], Cy = th[5];
  const float fx = (float)ox, fy = (float)oy;
  const float ix = Ax * fx + Bx * fy + Cx;
  const float iy = Ay * fx + By * fy + Cy;
  const float x0f = floorf(ix), y0f = floorf(iy);
  const float wx = ix - x0f, wy = iy - y0f;
  const int x0 = (int)x0f, y0 = (int)y0f;
  const float* P = up + c * WUP * WUP;
  const float v = smp_up(P, y0, x0) * (1.f - wx) * (1.f - wy) +
                  smp_up(P, y0, x0 + 1) * wx * (1.f - wy) +
                  smp_up(P, y0 + 1, x0) * (1.f - wx) * wy +
                  smp_up(P, y0 + 1, x0 + 1) * wx * wy;
  gs[idx] = v;
}

// ---------------- downsample x2 (crop -1 fused), 16x16 tile via WMMA -------
__device__ inline float gload(const float* __restrict__ g, int row, int col) {
  // row,col >= 1 by construction; only the upper bound can be exceeded, and
  // those taps have zero band coefficients -> clamp the gather, zero the value.
  const bool ok = (row < WO) & (col < WO);
  const int rc = row > WO - 1 ? WO - 1 : row;
  const int cc = col > WO - 1 ? WO - 1 : col;
  const float v = g[rc * WO + cc];
  return ok ? v : 0.f;
}

__global__ void __launch_bounds__(32)
downsample_kernel(const float* __restrict__ gs, const float* __restrict__ ws,
                  float* __restrict__ out) {
  const int lane = threadIdx.x, lo = lane & 15, hi = lane >> 4;
  const int c = blockIdx.z;
  const float* G = gs + c * WO * WO;
  const float* WdA = ws + OFF_WDA;
  const float* WdB = ws + OFF_WDB;
  const int Y0 = blockIdx.y * 16, X0 = blockIdx.x * 16;
  const int rowb = 2 * Y0 + 1, colb = 2 * X0 + 1;

  __builtin_prefetch(G + rowb * WO + colb, 0, 0);

  __shared__ float V[16][49];
  for (int cc = 0; cc < 3; cc++) {
    v8f acc = {0.f, 0.f, 0.f, 0.f, 0.f, 0.f, 0.f, 0.f};
#pragma unroll
    for (int kk = 0; kk < 12; kk++) {
      const int k0 = 4 * kk + 2 * hi;
      v2f a, b;
      a.x = WdA[lo * 48 + k0];
      a.y = WdA[lo * 48 + k0 + 1];
      const int col = colb + cc * 16 + lo;
      b.x = gload(G, rowb + k0, col);
      b.y = gload(G, rowb + k0 + 1, col);
      acc = __builtin_amdgcn_wmma_f32_16x16x4_f32(false, a, false, b, (short)0, acc, false, false);
    }
#pragma unroll
    for (int v = 0; v < 8; v++) V[v + 8 * hi][cc * 16 + lo] = acc[v];
  }
  __syncthreads();

  v8f o = {0.f, 0.f, 0.f, 0.f, 0.f, 0.f, 0.f, 0.f};
#pragma unroll
  for (int kk = 0; kk < 12; kk++) {
    const int k0 = 4 * kk + 2 * hi;
    v2f a, b;
    a.x = V[lo][k0];
    a.y = V[lo][k0 + 1];
    b.x = WdB[k0 * 16 + lo];
    b.y = WdB[(k0 + 1) * 16 + lo];
    o = __builtin_amdgcn_wmma_f32_16x16x4_f32(false, a, false, b, (short)0, o, false, false);
  }
#pragma unroll
  for (int v = 0; v < 8; v++)
    out[c * WIN * WIN + (Y0 + v + 8 * hi) * WIN + X0 + lo] = o[v];
}

// ---------------- host-side launcher ---------------------------------------
extern "C" void kernel_launch(void* const* d_in, const int* in_sizes, int n_in,
                              void* d_out, int out_size, void* d_ws, size_t ws_size,
                              hipStream_t stream) {
  (void)in_sizes; (void)n_in; (void)out_size; (void)ws_size;
  const float* images  = (const float*)d_in[0];
  const float* u_flip  = (const float*)d_in[1];
  const float* u_rot90 = (const float*)d_in[2];
  const float* u_int   = (const float*)d_in[3];
  const float* z_scale = (const float*)d_in[4];
  const float* u_rot1  = (const float*)d_in[5];
  const float* z_aniso = (const float*)d_in[6];
  const float* u_rot2  = (const float*)d_in[7];
  const float* z_frac  = (const float*)d_in[8];
  const float* hz      = (const float*)d_in[9];
  float* ws  = (float*)d_ws;
  float* outp = (float*)d_out;

  init_kernel<<<1, 256, 0, stream>>>(u_flip, u_rot90, u_int, z_scale, u_rot1, z_aniso,
                                     u_rot2, z_frac, hz, ws);

  float* up  = ws + OFF_UP;
  float* gsb = ws + OFF_GS;
  const int tot = 3 * WO * WO;
  for (int n = 0; n < 8; n++) {
    upsample_kernel<<<dim3(96, 96, 3), 32, 0, stream>>>(
        images + (size_t)n * 3 * WIN * WIN, ws, up);
    gridsample_kernel<<<dim3((tot + 255) / 256), 256, 0, stream>>>(
        up, ws + OFF_THETA + n * 6, gsb);
    downsample_kernel<<<dim3(16, 16, 3), 32, 0, stream>>>(
        gsb, ws, outp + (size_t)n * 3 * WIN * WIN);
  }
}